// InverseGNN_60103772340704
// MI455X (gfx1250) — compile-verified
//
#include <hip/hip_runtime.h>
#include <hip/hip_bf16.h>
#include <math.h>
#include <stdint.h>

// ---------------- constants (from reference) ----------------
#define NN 50000
#define EE 800000
#define BB 64
#define NODE_IN 16
#define EDGE_IN 8
#define HH 64
#define OUT_DIM 2

typedef __attribute__((ext_vector_type(2))) float v2f;
typedef __attribute__((ext_vector_type(8))) float v8f;

#if defined(__AMDGCN__) && __has_builtin(__builtin_amdgcn_global_load_async_to_lds_b128)
#define USE_ASYNC_LDS 1
typedef int v4i_vs __attribute__((vector_size(16)));
typedef __attribute__((address_space(1))) v4i_vs* g_v4i_p;   // global (AS1)
typedef __attribute__((address_space(3))) v4i_vs* l_v4i_p;   // LDS (AS3)
#else
#define USE_ASYNC_LDS 0
#endif

// monotone float <-> uint mapping for atomic float max
__device__ __forceinline__ unsigned fenc(float f) {
    unsigned u = __float_as_uint(f);
    return u ^ ((u >> 31) ? 0xFFFFFFFFu : 0x80000000u);
}
__device__ __forceinline__ float fdec(unsigned u) {
    u ^= ((u >> 31) ? 0x80000000u : 0xFFFFFFFFu);
    return __uint_as_float(u);
}

// ---------------------------------------------------------------------------
// Generic fp32 GEMM + bias: C[M,Nc] = A[M,K] @ W[K,Nc] + bias[Nc]
// block = 128 (4 waves); each wave computes one 16x16 tile of C via
// V_WMMA_F32_16X16X4_F32. grid = (M/16, Nc/64). K multiple of 4.
// ---------------------------------------------------------------------------
__global__ void gemm_bias_kernel(const float* __restrict__ A,
                                 const float* __restrict__ W,
                                 const float* __restrict__ bias,
                                 float* __restrict__ C,
                                 int M, int K, int Nc) {
    const int tid  = threadIdx.x;
    const int lane = tid & 31, wave = tid >> 5;
    const int half = lane >> 4;       // 0: lanes 0-15, 1: lanes 16-31
    const int l16  = lane & 15;
    const int row  = blockIdx.x * 16 + l16;               // A row held by lane
    const int col  = blockIdx.y * 64 + wave * 16 + l16;   // C column held by lane

    v8f c = {};
    for (int kb = 0; kb < K; kb += 4) {
        const int ka = kb + half * 2;         // A VGPR0 holds K{0,2}, VGPR1 K{1,3}
        v2f a, b;
        a.x = A[(size_t)row * K + ka];
        a.y = A[(size_t)row * K + ka + 1];
        b.x = W[(size_t)(ka)     * Nc + col];
        b.y = W[(size_t)(ka + 1) * Nc + col];
        c = __builtin_amdgcn_wmma_f32_16x16x4_f32(false, a, false, b,
                                                  (short)0, c, false, false);
    }
    const float bc = bias ? bias[col] : 0.0f;
    const int m0 = blockIdx.x * 16 + half * 8;   // D: VGPR v holds M = v + 8*half
#pragma unroll
    for (int v = 0; v < 8; ++v)
        C[(size_t)(m0 + v) * Nc + col] = c[v] + bc;
}

// ---------------------------------------------------------------------------
// Self-loop edge-attr fill: loop_attr[d] += ea[e], deg[d] += 1  (segment sum)
// ---------------------------------------------------------------------------
__global__ void loop_acc_kernel(const float* __restrict__ ea,
                                const long long* __restrict__ dst,
                                float* __restrict__ loopattr,
                                float* __restrict__ deg) {
    size_t idx = (size_t)blockIdx.x * blockDim.x + threadIdx.x;
    if (idx >= (size_t)EE * HH) return;
    int e = (int)(idx >> 6), k = (int)(idx & 63);
    int d = (int)dst[e];
    atomicAdd(&loopattr[(size_t)d * HH + k], ea[idx]);
    if (k == 0) atomicAdd(&deg[d], 1.0f);
}

__global__ void loop_div_kernel(float* __restrict__ loopattr,
                                const float* __restrict__ deg) {
    size_t idx = (size_t)blockIdx.x * blockDim.x + threadIdx.x;
    if (idx >= (size_t)NN * HH) return;
    float dg = deg[idx >> 6];
    loopattr[idx] /= (dg > 1.0f ? dg : 1.0f);
}

// ---------------------------------------------------------------------------
// Fused GATv2 edge pass 1: per 16-edge tile, gather edge features into LDS
// (async-to-LDS B128 when available), compute ep = eattr @ We with WMMA
// (ep lives only in LDS, never in HBM), then leaky-relu message + attention
// logits and segment-max via monotone-uint atomicMax.
// block = 256 threads (8 waves); grid = (E+N)/16.
// ---------------------------------------------------------------------------
__global__ void edge_logits_kernel(const float* __restrict__ ea,        // [E,64]
                                   const float* __restrict__ loopattr,  // [N,64]
                                   const long long* __restrict__ src,
                                   const long long* __restrict__ dst,
                                   const float* __restrict__ We,        // [64,F4]
                                   const float* __restrict__ att,       // [heads,64]
                                   const float* __restrict__ xl,        // [N,F4]
                                   const float* __restrict__ xr,        // [N,F4]
                                   float* __restrict__ logits,          // [E+N,heads]
                                   unsigned* __restrict__ lmax,         // [N,heads]
                                   int heads) {
    const int F4 = heads * HH;
    __shared__ float Atile[16 * HH];     // 4 KB  : gathered edge features
    __shared__ float eps[16 * 256];      // 16 KB : ep tile (worst case heads=4)
    __shared__ int   sI[16], dI[16];

    const int tid = threadIdx.x;
    const int e0  = blockIdx.x * 16;

    if (tid < 16) {
        int e = e0 + tid;
        if (e < EE) { sI[tid] = (int)src[e]; dI[tid] = (int)dst[e]; }
        else        { sI[tid] = dI[tid] = e - EE; }       // self-loop
    }
#if USE_ASYNC_LDS
    {   // 256 lanes x B128 = the whole 16x64 f32 tile, one async op per wave
        const int r = tid >> 4, k4 = (tid & 15) * 4;
        const int e = e0 + r;
        const float* gsrc = (e < EE) ? (ea + (size_t)e * HH + k4)
                                     : (loopattr + (size_t)(e - EE) * HH + k4);
        __builtin_amdgcn_global_load_async_to_lds_b128(
            (g_v4i_p)(uintptr_t)gsrc,
            (l_v4i_p)(uintptr_t)(&Atile[r * HH + k4]), 0, 0);
    }
#if __has_builtin(__builtin_amdgcn_s_wait_asynccnt)
    __builtin_amdgcn_s_wait_asynccnt(0);
#else
    asm volatile("s_wait_asynccnt 0x0" ::: "memory");
#endif
#else
    for (int i = tid; i < 16 * HH; i += 256) {
        int r = i >> 6, k = i & 63;
        int e = e0 + r;
        Atile[i] = (e < EE) ? ea[(size_t)e * HH + k]
                            : loopattr[(size_t)(e - EE) * HH + k];
    }
#endif
    __syncthreads();

    const int lane = tid & 31, wave = tid >> 5;
    const int half = lane >> 4, l16 = lane & 15;
    const int nct = F4 >> 4;
    for (int ct = wave; ct < nct; ct += 8) {
        v8f c = {};
        for (int kb = 0; kb < HH; kb += 4) {
            const int ka = kb + half * 2;
            v2f a, b;
            a.x = Atile[l16 * HH + ka];
            a.y = Atile[l16 * HH + ka + 1];
            b.x = We[(size_t)(ka)     * F4 + ct * 16 + l16];
            b.y = We[(size_t)(ka + 1) * F4 + ct * 16 + l16];
            c = __builtin_amdgcn_wmma_f32_16x16x4_f32(false, a, false, b,
                                                      (short)0, c, false, false);
        }
#pragma unroll
        for (int v = 0; v < 8; ++v)
            eps[(v + half * 8) * F4 + ct * 16 + l16] = c[v];
    }
    __syncthreads();

    if (tid < 16 * heads) {
        const int r = tid / heads, hh = tid % heads;
        const int s = sI[r], d = dI[r];
        const float* xls = xl  + (size_t)s * F4 + hh * HH;
        const float* xrd = xr  + (size_t)d * F4 + hh * HH;
        const float* ar  = att + hh * HH;
        const float* er  = eps + r * F4 + hh * HH;
        float acc = 0.0f;
        for (int cch = 0; cch < HH; ++cch) {
            float m = xls[cch] + xrd[cch] + er[cch];
            m = (m > 0.0f) ? m : 0.2f * m;               // leaky_relu(0.2)
            acc += m * ar[cch];
        }
        logits[(size_t)(e0 + r) * heads + hh] = acc;
        atomicMax(&lmax[(size_t)d * heads + hh], fenc(acc));
    }
}

// pass 2: a = exp(logit - lmax[d]); denom[d] += a  (stored over logits)
__global__ void alpha_kernel(float* __restrict__ logits,
                             const unsigned* __restrict__ lmax,
                             float* __restrict__ denom,
                             const long long* __restrict__ dst, int heads) {
    size_t total = (size_t)(EE + NN) * heads;
    size_t idx = (size_t)blockIdx.x * blockDim.x + threadIdx.x;
    if (idx >= total) return;
    int e = (int)(idx / heads), hh = (int)(idx % heads);
    int d = (e < EE) ? (int)dst[e] : e - EE;
    float a = __expf(logits[idx] - fdec(lmax[(size_t)d * heads + hh]));
    logits[idx] = a;
    atomicAdd(&denom[(size_t)d * heads + hh], a);
}

// pass 3: out[d] += (a / denom[d]) * xl[s]  — 4 channels per thread (float4)
__global__ void aggregate_kernel(const float* __restrict__ logits,
                                 const float* __restrict__ denom,
                                 const long long* __restrict__ src,
                                 const long long* __restrict__ dst,
                                 const float* __restrict__ xl,
                                 float* __restrict__ out, int heads) {
    const int F4 = heads * HH;
    const int Q  = F4 >> 2;                       // float4 groups per edge
    size_t total = (size_t)(EE + NN) * Q;
    size_t idx = (size_t)blockIdx.x * blockDim.x + threadIdx.x;
    if (idx >= total) return;
    int e  = (int)(idx / Q);
    int fc = (int)(idx % Q) * 4;
    int hh = fc >> 6;
    int s = (e < EE) ? (int)src[e] : e - EE;
    int d = (e < EE) ? (int)dst[e] : e - EE;
    float a = logits[(size_t)e * heads + hh];
    float w = a / (denom[(size_t)d * heads + hh] + 1e-16f);
    const float4 v = *(const float4*)(xl + (size_t)s * F4 + fc);
    float* op = out + (size_t)d * F4 + fc;
    atomicAdd(op + 0, w * v.x);
    atomicAdd(op + 1, w * v.y);
    atomicAdd(op + 2, w * v.z);
    atomicAdd(op + 3, w * v.w);
}

// h1 = elu(agg + bias)  (in place), length NN*F4
__global__ void elu_bias_kernel(float* __restrict__ agg,
                                const float* __restrict__ bias, int F4) {
    size_t idx = (size_t)blockIdx.x * blockDim.x + threadIdx.x;
    if (idx >= (size_t)NN * F4) return;
    float v = agg[idx] + bias[idx % F4];
    agg[idx] = (v > 0.0f) ? v : (__expf(v) - 1.0f);
}

// h2 = agg + bias (in place), length NN*F4
__global__ void add_bias_kernel(float* __restrict__ agg,
                                const float* __restrict__ bias, int F4) {
    size_t idx = (size_t)blockIdx.x * blockDim.x + threadIdx.x;
    if (idx >= (size_t)NN * F4) return;
    agg[idx] += bias[idx % F4];
}

// global mean-pool accumulation
__global__ void pool_kernel(const float* __restrict__ h2,
                            const long long* __restrict__ batch,
                            float* __restrict__ pooled, float* __restrict__ cnt) {
    size_t idx = (size_t)blockIdx.x * blockDim.x + threadIdx.x;
    if (idx >= (size_t)NN * HH) return;
    int n = (int)(idx >> 6), c = (int)(idx & 63);
    int b = (int)batch[n];
    atomicAdd(&pooled[(size_t)b * HH + c], h2[idx]);
    if (c == 0) atomicAdd(&cnt[b], 1.0f);
}

// decoder: sigmoid(relu(pooled/cnt @ Wd1 + bd1) @ Wd2 + bd2)  — one block, 64 thr
__global__ void decoder_kernel(const float* __restrict__ pooled,
                               const float* __restrict__ cnt,
                               const float* __restrict__ Wd1,
                               const float* __restrict__ bd1,
                               const float* __restrict__ Wd2,
                               const float* __restrict__ bd2,
                               float* __restrict__ out) {
    __shared__ float p[BB * HH];
    __shared__ float t[BB * HH];
    const int b = threadIdx.x;           // 0..63, one batch row per thread
    float c = cnt[b]; c = (c > 1.0f) ? c : 1.0f;
    for (int k = 0; k < HH; ++k) p[b * HH + k] = pooled[b * HH + k] / c;
    __syncthreads();
    for (int j = 0; j < HH; ++j) {
        float acc = bd1[j];
        for (int k = 0; k < HH; ++k) acc += p[b * HH + k] * Wd1[k * HH + j];
        t[b * HH + j] = (acc > 0.0f) ? acc : 0.0f;
    }
    __syncthreads();
    for (int o = 0; o < OUT_DIM; ++o) {
        float acc = bd2[o];
        for (int j = 0; j < HH; ++j) acc += t[b * HH + j] * Wd2[j * OUT_DIM + o];
        out[b * OUT_DIM + o] = 1.0f / (1.0f + __expf(-acc));
    }
}

// ---------------------------------------------------------------------------
extern "C" void kernel_launch(void* const* d_in, const int* in_sizes, int n_in,
                              void* d_out, int out_size, void* d_ws, size_t ws_size,
                              hipStream_t stream) {
    const float*     x      = (const float*)d_in[0];
    const long long* eidx   = (const long long*)d_in[1];   // [2,E] int64
    const float*     eattr  = (const float*)d_in[2];
    const long long* batch  = (const long long*)d_in[3];
    const float* W_node = (const float*)d_in[4];  const float* b_node = (const float*)d_in[5];
    const float* W_edge = (const float*)d_in[6];  const float* b_edge = (const float*)d_in[7];
    const float* Wl1 = (const float*)d_in[8];  const float* bl1 = (const float*)d_in[9];
    const float* Wr1 = (const float*)d_in[10]; const float* br1 = (const float*)d_in[11];
    const float* We1 = (const float*)d_in[12]; const float* att1 = (const float*)d_in[13];
    const float* bias1 = (const float*)d_in[14];
    const float* Wl2 = (const float*)d_in[15]; const float* bl2 = (const float*)d_in[16];
    const float* Wr2 = (const float*)d_in[17]; const float* br2 = (const float*)d_in[18];
    const float* We2 = (const float*)d_in[19]; const float* att2 = (const float*)d_in[20];
    const float* bias2 = (const float*)d_in[21];
    const float* Wd1 = (const float*)d_in[22]; const float* bd1 = (const float*)d_in[23];
    const float* Wd2 = (const float*)d_in[24]; const float* bd2 = (const float*)d_in[25];
    float* out = (float*)d_out;

    const long long* src = eidx;
    const long long* dst = eidx + EE;

    // ---- workspace layout (floats) ----
    float* ws = (float*)d_ws;
    size_t o = 0;
    float* ea      = ws + o; o += (size_t)EE * HH;        // encoded edge attrs
    float* loopat  = ws + o; o += (size_t)NN * HH;        // self-loop attrs
    float* deg     = ws + o; o += NN;
    float* hbuf    = ws + o; o += (size_t)NN * HH;        // enc nodes / conv2 agg / h2
    float* xl      = ws + o; o += (size_t)NN * 4 * HH;    // xl1 (xl2 reuses prefix)
    float* xr      = ws + o; o += (size_t)NN * 4 * HH;    // xr1 (xr2 reuses prefix)
    float* agg     = ws + o; o += (size_t)NN * 4 * HH;    // conv1 agg -> h1
    float* logits  = ws + o; o += (size_t)(EE + NN) * 4;
    unsigned* lmax = (unsigned*)(ws + o); o += (size_t)NN * 4;
    float* denom   = ws + o; o += (size_t)NN * 4;
    float* pooled  = ws + o; o += (size_t)BB * HH;
    float* cnt     = ws + o; o += BB;
    (void)ws_size; (void)n_in; (void)in_sizes; (void)out_size;

    const dim3 b128(128), b256(256);
    auto g1 = [](size_t total) { return dim3((unsigned)((total + 255) / 256)); };

    // ---- encoders ----
    gemm_bias_kernel<<<dim3(NN/16, HH/64), b128, 0, stream>>>(x, W_node, b_node, hbuf, NN, NODE_IN, HH);
    gemm_bias_kernel<<<dim3(EE/16, HH/64), b128, 0, stream>>>(eattr, W_edge, b_edge, ea, EE, EDGE_IN, HH);

    // ---- self-loop attrs (shared by both convs) ----
    (void)hipMemsetAsync(loopat, 0, (size_t)NN * HH * sizeof(float), stream);
    (void)hipMemsetAsync(deg,    0, (size_t)NN * sizeof(float), stream);
    loop_acc_kernel<<<g1((size_t)EE * HH), b256, 0, stream>>>(ea, dst, loopat, deg);
    loop_div_kernel<<<g1((size_t)NN * HH), b256, 0, stream>>>(loopat, deg);

    // ---- conv1 (heads=4, concat) ----
    gemm_bias_kernel<<<dim3(NN/16, (4*HH)/64), b128, 0, stream>>>(hbuf, Wl1, bl1, xl, NN, HH, 4*HH);
    gemm_bias_kernel<<<dim3(NN/16, (4*HH)/64), b128, 0, stream>>>(hbuf, Wr1, br1, xr, NN, HH, 4*HH);
    (void)hipMemsetAsync(lmax,  0, (size_t)NN * 4 * sizeof(unsigned), stream);
    (void)hipMemsetAsync(denom, 0, (size_t)NN * 4 * sizeof(float), stream);
    (void)hipMemsetAsync(agg,   0, (size_t)NN * 4 * HH * sizeof(float), stream);
    edge_logits_kernel<<<dim3((EE+NN)/16), b256, 0, stream>>>(ea, loopat, src, dst, We1, att1,
                                                              xl, xr, logits, lmax, 4);
    alpha_kernel<<<g1((size_t)(EE+NN) * 4), b256, 0, stream>>>(logits, lmax, denom, dst, 4);
    aggregate_kernel<<<g1((size_t)(EE+NN) * 64), b256, 0, stream>>>(logits, denom, src, dst,
                                                                    xl, agg, 4);
    elu_bias_kernel<<<g1((size_t)NN * 4 * HH), b256, 0, stream>>>(agg, bias1, 4*HH);  // agg = h1

    // ---- conv2 (heads=1, mean==identity) ----
    gemm_bias_kernel<<<dim3(NN/16, HH/64), b128, 0, stream>>>(agg, Wl2, bl2, xl, NN, 4*HH, HH);
    gemm_bias_kernel<<<dim3(NN/16, HH/64), b128, 0, stream>>>(agg, Wr2, br2, xr, NN, 4*HH, HH);
    (void)hipMemsetAsync(lmax,  0, (size_t)NN * sizeof(unsigned), stream);
    (void)hipMemsetAsync(denom, 0, (size_t)NN * sizeof(float), stream);
    (void)hipMemsetAsync(hbuf,  0, (size_t)NN * HH * sizeof(float), stream);
    edge_logits_kernel<<<dim3((EE+NN)/16), b256, 0, stream>>>(ea, loopat, src, dst, We2, att2,
                                                              xl, xr, logits, lmax, 1);
    alpha_kernel<<<g1((size_t)(EE+NN)), b256, 0, stream>>>(logits, lmax, denom, dst, 1);
    aggregate_kernel<<<g1((size_t)(EE+NN) * 16), b256, 0, stream>>>(logits, denom, src, dst,
                                                                    xl, hbuf, 1);
    add_bias_kernel<<<g1((size_t)NN * HH), b256, 0, stream>>>(hbuf, bias2, HH);  // hbuf = h2

    // ---- global mean pool + decoder ----
    (void)hipMemsetAsync(pooled, 0, (size_t)BB * HH * sizeof(float), stream);
    (void)hipMemsetAsync(cnt,    0, (size_t)BB * sizeof(float), stream);
    pool_kernel<<<g1((size_t)NN * HH), b256, 0, stream>>>(hbuf, batch, pooled, cnt);
    decoder_kernel<<<dim3(1), dim3(BB), 0, stream>>>(pooled, cnt, Wd1, bd1, Wd2, bd2, out);
}